// S4DKernel_14139032339101
// MI455X (gfx1250) — compile-verified
//
#include <hip/hip_runtime.h>

// S4D Vandermonde kernel for gfx1250 (MI455X), FP32 WMMA path.
// out[h, 16*lo+li] = sum_k A[lo,k]*B[k,li] with K=64 (real/imag split),
// computed per-head as (512x64)@(64x16) via chained v_wmma_f32_16x16x4_f32.

typedef __attribute__((ext_vector_type(2))) float v2f;
typedef __attribute__((ext_vector_type(8))) float v8f;

#define S4D_H    1024
#define S4D_N    32      // N/2 complex modes
#define S4D_L    8192
#define TILE_LI  16      // inner l factor (WMMA N dim)
#define KDIM     64      // 2 * S4D_N (re/im split)

__global__ __launch_bounds__(256)
void s4d_vandermonde_wmma(const float* __restrict__ Cp,         // (H, 32, 2)
                          const float* __restrict__ log_dt,     // (H,)
                          const float* __restrict__ log_A_real, // (H, 32)
                          const float* __restrict__ A_imag,     // (H, 32)
                          float* __restrict__ out)              // (H, L)
{
    __shared__ float2 sdtA[S4D_N];
    __shared__ float2 sCeff[S4D_N];
    __shared__ float  sB[KDIM * TILE_LI];   // 64x16 f32 = 4 KB

    const int h   = blockIdx.x;
    const int tid = threadIdx.x;

    // ---- per-head parameters: dtA and discretized Ceff (x2 folded in) ----
    if (tid < S4D_N) {
        const int n = tid;
        float dt = __expf(log_dt[h]);
        float ar = -__expf(log_A_real[h * S4D_N + n]);
        float ai = A_imag[h * S4D_N + n];
        float dx = ar * dt;                 // Re(dtA)
        float dy = ai * dt;                 // Im(dtA)
        // z = exp(dtA) - 1
        float e  = __expf(dx);
        float sn, cs; __sincosf(dy, &sn, &cs);
        float zr = e * cs - 1.0f;
        float zi = e * sn;
        // w = z / A,  A = ar + i*ai
        float inv = 1.0f / (ar * ar + ai * ai);
        float wr = (zr * ar + zi * ai) * inv;
        float wi = (zi * ar - zr * ai) * inv;
        // Ceff = 2 * Cc * w
        float cr = Cp[(h * S4D_N + n) * 2 + 0];
        float ci = Cp[(h * S4D_N + n) * 2 + 1];
        sCeff[n] = make_float2(2.0f * (cr * wr - ci * wi),
                               2.0f * (cr * wi + ci * wr));
        sdtA[n]  = make_float2(dx, dy);
    }
    __syncthreads();

    // ---- shared B tile: B[k][li] = Re/Im(exp(dtA_n * li)), k<32 -> Re ----
    for (int idx = tid; idx < KDIM * TILE_LI; idx += 256) {
        int k  = idx >> 4;
        int li = idx & 15;
        int n  = k & (S4D_N - 1);
        float2 d = sdtA[n];
        float t  = (float)li;
        float e  = __expf(d.x * t);
        float sn, cs; __sincosf(d.y * t, &sn, &cs);
        sB[idx] = (k < S4D_N) ? (e * cs) : (e * sn);
    }
    __syncthreads();

    const int lane = tid & 31;
    const int wave = tid >> 5;
    const int col  = lane & 15;            // N index / A-row M index
    const int kb   = (lane >> 4) << 1;     // K sub-pair: 0 for lanes 0-15, 2 for 16-31

    // ---- B fragments, resident for all tiles of this head ----
    // B 4x16 f32 layout: lanes 0-15 hold rows k0+0/k0+1, lanes 16-31 rows k0+2/k0+3.
    v2f bfrag[16];
    #pragma unroll
    for (int c = 0; c < 16; ++c) {
        int row = 4 * c + kb;
        bfrag[c][0] = sB[(row + 0) * TILE_LI + col];
        bfrag[c][1] = sB[(row + 1) * TILE_LI + col];
    }

    // ---- each wave computes 4 lo-tiles of 16 rows ----
    #pragma unroll
    for (int t = 0; t < 4; ++t) {
        const int   lo_base = (wave * 4 + t) * 16;
        const int   lo      = lo_base + col;          // this lane's A-matrix row M
        const float tl      = (float)(lo * TILE_LI);  // 16*lo

        // A 16x4 f32 layout: lanes 0-15 hold K=k0+0/k0+1, lanes 16-31 K=k0+2/k0+3.
        // Chunks c<8 carry Re(Ceff*P), chunks c>=8 carry -Im(Ceff*P) (same n).
        v2f afrag[16];
        #pragma unroll
        for (int c = 0; c < 8; ++c) {
            #pragma unroll
            for (int r = 0; r < 2; ++r) {
                int n = 4 * c + kb + r;
                float2 d  = sdtA[n];
                float2 ce = sCeff[n];
                float e  = __expf(d.x * tl);
                float sn, cs; __sincosf(d.y * tl, &sn, &cs);
                float pr = e * cs, pi = e * sn;
                afrag[c][r]     =  ce.x * pr - ce.y * pi;   //  Re
                afrag[c + 8][r] = -(ce.x * pi + ce.y * pr); // -Im
            }
        }

        v8f acc = {};
        #pragma unroll
        for (int c = 0; c < 16; ++c) {
            acc = __builtin_amdgcn_wmma_f32_16x16x4_f32(
                false, afrag[c], false, bfrag[c],
                (short)0, acc, false, false);
        }

        // D layout: VGPR r -> row r (lanes 0-15) / row r+8 (lanes 16-31).
        float* op = out + (size_t)h * S4D_L + (size_t)lo_base * TILE_LI;
        const int mhalf = (lane >> 4) * 8;
        #pragma unroll
        for (int r = 0; r < 8; ++r) {
            op[(mhalf + r) * TILE_LI + col] = acc[r];
        }
    }
}

extern "C" void kernel_launch(void* const* d_in, const int* in_sizes, int n_in,
                              void* d_out, int out_size, void* d_ws, size_t ws_size,
                              hipStream_t stream) {
    (void)in_sizes; (void)n_in; (void)out_size; (void)d_ws; (void)ws_size;
    const float* C          = (const float*)d_in[0]; // (H, 32, 2)
    const float* log_dt     = (const float*)d_in[1]; // (H,)
    const float* log_A_real = (const float*)d_in[2]; // (H, 32)
    const float* A_imag     = (const float*)d_in[3]; // (H, 32)
    // d_in[4] is the Python scalar L (== 8192), baked in as S4D_L.
    float* out = (float*)d_out;                      // (H, L) f32

    s4d_vandermonde_wmma<<<S4D_H, 256, 0, stream>>>(C, log_dt, log_A_real,
                                                    A_imag, out);
}